// MetaProperty_42236708389807
// MI455X (gfx1250) — compile-verified
//
#include <hip/hip_runtime.h>

#define B_SZ   256
#define NU_ROWS 47155
#define NI_ROWS 16532
#define OUT_DIM 600
#define KTAPS  754
#define KFULL  30160      // 40*754
#define REV_STRIDE 30720  // 40*768

#define KSPLIT 8          // K-slices (channels split 40 -> 8 x 5)
#define CH_PER_SLICE 5
#define TCHUNK 64         // taps staged per LDS chunk
#define NCHUNK 12         // 12 chunks cover 768 taps (zero padded past 754)
#define LDS_STRIDE 68     // 68 mod 64 == 4 -> conflict-free b64 reads
#define FEAT_N (B_SZ * OUT_DIM)   // 153600 floats per side

typedef __attribute__((ext_vector_type(2))) float v2f;
typedef __attribute__((ext_vector_type(8))) float v8f;

// ---------------------------------------------------------------------------
// Kernel 1: w[b,i] = (1/6) * sum_p scores[b,p,i] * pref_table[id[b], p]
// ---------------------------------------------------------------------------
__global__ __launch_bounds__(256) void weights_kernel(
    const int* __restrict__ uid, const int* __restrict__ iid,
    const float* __restrict__ uscores, const float* __restrict__ iscores,
    const float* __restrict__ upref, const float* __restrict__ ipref,
    float* __restrict__ w_u, float* __restrict__ w_i)
{
    int idx = blockIdx.x * blockDim.x + threadIdx.x;
    if (idx >= B_SZ * 40) return;
    int b = idx / 40, i = idx % 40;

    int u = uid[b];
    float su = 0.f;
    #pragma unroll
    for (int p = 0; p < 6; ++p)
        su += uscores[(b * 6 + p) * 40 + i] * upref[u * 6 + p];
    w_u[idx] = su * (1.f / 6.f);

    int it = iid[b];
    float si = 0.f;
    #pragma unroll
    for (int p = 0; p < 6; ++p)
        si += iscores[(b * 6 + p) * 40 + i] * ipref[it * 6 + p];
    w_i[idx] = si * (1.f / 6.f);
}

// ---------------------------------------------------------------------------
// Kernel 2: conv-as-GEMM via V_WMMA_F32_16X16X4_F32, K split 8 ways.
//   M = 3840 rows (b,j), N = 40 cols (o), K-slice = 5 channels x 754 taps.
//   B chunk (48 o x 64 t) staged in LDS (float2 granularity) once per block,
//   shared by 8 waves. Partials -> part[side][kslice][b*600 + o*15 + j].
// ---------------------------------------------------------------------------
__global__ __launch_bounds__(256) void conv_gemm_kernel(
    const float* __restrict__ urev, const float* __restrict__ irev,
    const float* __restrict__ w_u, const float* __restrict__ w_i,
    const float* __restrict__ convw_u, const float* __restrict__ convw_i,
    float* __restrict__ part)
{
    __shared__ float smB[48 * LDS_STRIDE];   // 13056 B

    const int side   = blockIdx.y;
    const int kslice = blockIdx.z;
    const float* __restrict__ review = side ? irev    : urev;
    const float* __restrict__ wtab   = side ? w_i     : w_u;
    const float* __restrict__ convw  = side ? convw_i : convw_u;

    const int tid  = threadIdx.x;
    const int wave = tid >> 5;
    const int lane = tid & 31;
    const int m    = lane & 15;     // row (A/M) or col (B/N) within tile
    const int kq   = lane >> 4;     // which K-pair this half-wave holds

    const int mtile = blockIdx.x * 8 + wave;   // 0..239
    const int row   = mtile * 16 + m;          // 0..3839
    const int b     = row / 15;
    const int j     = row % 15;

    const float* __restrict__ revB = review + (size_t)b * REV_STRIDE;
    const float* __restrict__ wB   = wtab + b * 40;

    const int o0 = m;
    const int o1 = 16 + m;
    const int o2 = 32 + m;          // valid only if < 40

    // staging decomposition: pair index -> (o row, tap pair) ; 1536 pairs
    const int st_o  = tid >> 5;          // pairs 0..31 of row (tid%32)*2 taps
    const int st_tt = (tid & 31) * 2;    // even tap within chunk

    v8f acc0 = {}, acc1 = {}, acc2 = {};

    for (int ii = 0; ii < CH_PER_SLICE; ++ii) {
        const int i = kslice * CH_PER_SLICE + ii;
        const float scale = wB[i];
        const float* __restrict__ aCh = revB + i * 768 + j + 2 * kq;
        const float* __restrict__ bCh = convw + (size_t)i * KTAPS;

        for (int tc = 0; tc < NCHUNK; ++tc) {
            const int tbase = tc * TCHUNK;

            __syncthreads();    // previous chunk fully consumed
            // ---- stage B chunk as float2: 48 rows x 32 pairs, 6 per thread.
            // Pair-granular masking is exact: KTAPS is even, so a pair at
            // even t is entirely valid (t<754) or entirely padding.
            #pragma unroll
            for (int e = 0; e < 6; ++e) {
                int o = st_o + e * 8;          // 0..47
                int t = tbase + st_tt;
                v2f v = {0.f, 0.f};
                if (o < 40 && t < KTAPS)
                    v = *(const v2f*)(bCh + (size_t)o * KFULL + t);
                *(v2f*)(smB + o * LDS_STRIDE + st_tt) = v;
            }
            __syncthreads();

            const float* aP = aCh + tbase;
            const float* s0 = smB + m * LDS_STRIDE + 2 * kq;
            const float* s1 = s0 + 16 * LDS_STRIDE;
            const float* s2 = s0 + 32 * LDS_STRIDE;

            if (tc < NCHUNK - 1) {
                #pragma unroll
                for (int t0 = 0; t0 < TCHUNK; t0 += 4) {
                    v2f a;
                    a.x = aP[t0] * scale;
                    a.y = aP[t0 + 1] * scale;
                    v2f f0 = *(const v2f*)(s0 + t0);
                    v2f f1 = *(const v2f*)(s1 + t0);
                    v2f f2 = *(const v2f*)(s2 + t0);
                    acc0 = __builtin_amdgcn_wmma_f32_16x16x4_f32(
                        false, a, false, f0, (short)0, acc0, false, false);
                    acc1 = __builtin_amdgcn_wmma_f32_16x16x4_f32(
                        false, a, false, f1, (short)0, acc1, false, false);
                    acc2 = __builtin_amdgcn_wmma_f32_16x16x4_f32(
                        false, a, false, f2, (short)0, acc2, false, false);
                }
            } else {
                // tail chunk: taps 704..767, only t<754 contribute (A masked;
                // B already zero-padded in LDS)
                #pragma unroll
                for (int t0 = 0; t0 < TCHUNK; t0 += 4) {
                    int t = tbase + t0 + 2 * kq;
                    v2f a;
                    a.x = (t     < KTAPS) ? aP[t0] * scale     : 0.f;
                    a.y = (t + 1 < KTAPS) ? aP[t0 + 1] * scale : 0.f;
                    v2f f0 = *(const v2f*)(s0 + t0);
                    v2f f1 = *(const v2f*)(s1 + t0);
                    v2f f2 = *(const v2f*)(s2 + t0);
                    acc0 = __builtin_amdgcn_wmma_f32_16x16x4_f32(
                        false, a, false, f0, (short)0, acc0, false, false);
                    acc1 = __builtin_amdgcn_wmma_f32_16x16x4_f32(
                        false, a, false, f1, (short)0, acc1, false, false);
                    acc2 = __builtin_amdgcn_wmma_f32_16x16x4_f32(
                        false, a, false, f2, (short)0, acc2, false, false);
                }
            }
        }
    }

    // ---- store raw partial sums (bias/relu applied in reduce kernel)
    float* __restrict__ dst0 =
        part + (size_t)(side * KSPLIT + kslice) * FEAT_N;
    #pragma unroll
    for (int r = 0; r < 8; ++r) {
        const int grow = mtile * 16 + r + 8 * kq;
        const int bb = grow / 15;
        const int jj = grow % 15;
        float* dst = dst0 + (size_t)bb * OUT_DIM + jj;
        dst[o0 * 15] = acc0[r];
        dst[o1 * 15] = acc1[r];
        if (o2 < 40)
            dst[o2 * 15] = acc2[r];
    }
}

// ---------------------------------------------------------------------------
// Kernel 3: feat = relu(sum_slices(part) + conv_b)   (deterministic reduce)
// ---------------------------------------------------------------------------
__global__ __launch_bounds__(256) void reduce_kernel(
    const float* __restrict__ part,
    const float* __restrict__ convb_u, const float* __restrict__ convb_i,
    float* __restrict__ feat_u, float* __restrict__ feat_i)
{
    int idx = blockIdx.x * blockDim.x + threadIdx.x;
    if (idx >= 2 * FEAT_N) return;
    int side = idx / FEAT_N;
    int x    = idx % FEAT_N;
    const float* p = part + (size_t)side * KSPLIT * FEAT_N + x;
    float s = 0.f;
    #pragma unroll
    for (int k = 0; k < KSPLIT; ++k)
        s += p[(size_t)k * FEAT_N];
    int o = (x % OUT_DIM) / 15;
    s += (side ? convb_i : convb_u)[o];
    (side ? feat_i : feat_u)[x] = fmaxf(s, 0.f);
}

// ---------------------------------------------------------------------------
// Kernel 4: scatter feature rows into the copied embedding tables
// ---------------------------------------------------------------------------
__global__ __launch_bounds__(256) void scatter_kernel(
    const int* __restrict__ uid, const int* __restrict__ iid,
    const float* __restrict__ feat_u, const float* __restrict__ feat_i,
    float* __restrict__ out_uemb, float* __restrict__ out_iemb)
{
    const int b = blockIdx.x;
    const int u  = uid[b];
    const int it = iid[b];
    for (int k = threadIdx.x; k < OUT_DIM; k += blockDim.x) {
        out_uemb[(size_t)u  * OUT_DIM + k] = feat_u[(size_t)b * OUT_DIM + k];
        out_iemb[(size_t)it * OUT_DIM + k] = feat_i[(size_t)b * OUT_DIM + k];
    }
}

// ---------------------------------------------------------------------------
// Kernel 5: out[b] = dot(feat_u[b], feat_i[b]) + biases + mu  (1 wave / row)
// ---------------------------------------------------------------------------
__global__ __launch_bounds__(32) void dot_kernel(
    const int* __restrict__ uid, const int* __restrict__ iid,
    const float* __restrict__ feat_u, const float* __restrict__ feat_i,
    const float* __restrict__ ubias, const float* __restrict__ ibias,
    const float* __restrict__ mu, float* __restrict__ out)
{
    const int b = blockIdx.x;
    const int lane = threadIdx.x;
    float s = 0.f;
    for (int k = lane; k < OUT_DIM; k += 32)
        s += feat_u[(size_t)b * OUT_DIM + k] * feat_i[(size_t)b * OUT_DIM + k];
    #pragma unroll
    for (int off = 16; off > 0; off >>= 1)
        s += __shfl_down(s, off, 32);
    if (lane == 0)
        out[b] = s + ubias[uid[b]] + ibias[iid[b]] + mu[0];
}

// ---------------------------------------------------------------------------
extern "C" void kernel_launch(void* const* d_in, const int* in_sizes, int n_in,
                              void* d_out, int out_size, void* d_ws, size_t ws_size,
                              hipStream_t stream)
{
    const int*   uid     = (const int*)  d_in[0];
    const int*   iid     = (const int*)  d_in[1];
    const float* urev    = (const float*)d_in[2];
    const float* uscores = (const float*)d_in[3];
    const float* irev    = (const float*)d_in[4];
    const float* iscores = (const float*)d_in[5];
    const float* upref   = (const float*)d_in[6];
    const float* ipref   = (const float*)d_in[7];
    const float* convw_u = (const float*)d_in[8];
    const float* convb_u = (const float*)d_in[9];
    const float* convw_i = (const float*)d_in[10];
    const float* convb_i = (const float*)d_in[11];
    const float* ubias   = (const float*)d_in[12];
    const float* ibias   = (const float*)d_in[13];
    const float* mu      = (const float*)d_in[14];
    const float* uemb    = (const float*)d_in[15];
    const float* iemb    = (const float*)d_in[16];

    float* out_scores = (float*)d_out;                        // [256]
    float* out_uemb   = out_scores + B_SZ;                    // [NU,600]
    float* out_iemb   = out_uemb + (size_t)NU_ROWS * OUT_DIM; // [NI,600]

    float* w_u    = (float*)d_ws;                 // [256,40]
    float* w_i    = w_u + B_SZ * 40;              // [256,40]
    float* feat_u = w_i + B_SZ * 40;              // [256,600]
    float* feat_i = feat_u + FEAT_N;              // [256,600]
    float* part   = feat_i + FEAT_N;              // [2][8][256,600] ~ 9.8 MB

    // 1) per-(b,i) attention weights
    weights_kernel<<<(B_SZ * 40 + 255) / 256, 256, 0, stream>>>(
        uid, iid, uscores, iscores, upref, ipref, w_u, w_i);

    // 2) WMMA GEMM: 30 mtile-groups x 2 sides x 8 K-slices = 480 blocks
    conv_gemm_kernel<<<dim3(30, 2, KSPLIT), 256, 0, stream>>>(
        urev, irev, w_u, w_i, convw_u, convw_i, part);

    // 3) copy embedding tables to output (graph-capture-safe D2D copies)
    hipMemcpyAsync(out_uemb, uemb, (size_t)NU_ROWS * OUT_DIM * sizeof(float),
                   hipMemcpyDeviceToDevice, stream);
    hipMemcpyAsync(out_iemb, iemb, (size_t)NI_ROWS * OUT_DIM * sizeof(float),
                   hipMemcpyDeviceToDevice, stream);

    // 4) deterministic K-slice reduction + bias + relu
    reduce_kernel<<<(2 * FEAT_N + 255) / 256, 256, 0, stream>>>(
        part, convb_u, convb_i, feat_u, feat_i);

    // 5) scatter feature rows over the copies
    scatter_kernel<<<B_SZ, 256, 0, stream>>>(uid, iid, feat_u, feat_i,
                                             out_uemb, out_iemb);

    // 6) final per-row dot + biases
    dot_kernel<<<B_SZ, 32, 0, stream>>>(uid, iid, feat_u, feat_i,
                                        ubias, ibias, mu, out_scores);
}